// EIGWhiteningSigma_Single_19963007992127
// MI455X (gfx1250) — compile-verified
//
#include <hip/hip_runtime.h>

// ZCA whitening via fp32 WMMA (V_WMMA_F32_16X16X4_F32) + Newton-Schulz inverse sqrt.
// X: [64, 256, 4096] fp32.  m = 64*4096 = 262144.

#define C_CH   256
#define L_LEN  4096
#define N_B    64
#define M_TOT  (N_B * L_LEN)
#define EPSV   1e-3f
#define NS_ITERS 40

typedef __attribute__((ext_vector_type(2))) float v2f;
typedef __attribute__((ext_vector_type(8))) float v8f;

__device__ __forceinline__ v8f wmma4(v2f a, v2f b, v8f c) {
  // D = A(16x4) x B(4x16) + C(16x16), all fp32.
  return __builtin_amdgcn_wmma_f32_16x16x4_f32(false, a, false, b, (short)0, c,
                                               false, false);
}

// ---------------------------------------------------------------- utilities
__global__ void zero_kernel(float* __restrict__ p, int n) {
  int i = blockIdx.x * blockDim.x + threadIdx.x;
  if (i < n) p[i] = 0.0f;
}

// Per-channel sums: SUM[c] = sum_{n,l} X[n,c,l]
__global__ void chan_sum_kernel(const float* __restrict__ X, float* __restrict__ SUM) {
  const int c = blockIdx.x;
  const int t = threadIdx.x;
  float s = 0.0f;
  for (int n = 0; n < N_B; ++n) {
    const float* row = X + ((size_t)(n * C_CH + c)) * L_LEN;
    for (int l = t; l < L_LEN; l += 256) s += row[l];
  }
  __shared__ float red[256];
  red[t] = s;
  __syncthreads();
  for (int off = 128; off > 0; off >>= 1) {
    if (t < off) red[t] += red[t + off];
    __syncthreads();
  }
  if (t == 0) SUM[c] = red[0];
}

// ------------------------------------------------- covariance (rank-m WMMA)
// Grid: x = 8 tile positions (2 row-tiles of 128 x 4 col-tiles of 64),
//       y = 64 k-chunks (one batch n each).
// Each of the 8 waves computes a 32x32 block (2x2 WMMA subtiles): 4 WMMAs
// per 2 paired-LDS fragment loads -> 0.5 DS instr / WMMA.
// LDS panels padded to stride 68 dwords (conflict-free fragment reads).
__global__ void cov_kernel(const float* __restrict__ X, float* __restrict__ S) {
  const int ti = blockIdx.x >> 2;   // 0..1  (128-row tile)
  const int tj = blockIdx.x & 3;    // 0..3  (64-col tile)
  const int n  = blockIdx.y;
  const int t = threadIdx.x, lane = t & 31, wv = t >> 5;
  const int half = lane >> 4, mn = lane & 15, k0 = half * 2;
  const int rg = wv >> 1;           // row group 0..3 (32 rows each)
  const int cg = wv & 1;            // col group 0..1 (32 cols each)

  __shared__ float pA[128 * 68];    // 34816 B
  __shared__ float pB[64 * 68];     // 17408 B

  v8f acc00 = {}, acc01 = {}, acc10 = {}, acc11 = {};

  const float* XA = X + ((size_t)(n * C_CH + ti * 128)) * L_LEN;
  const float* XB = X + ((size_t)(n * C_CH + tj * 64)) * L_LEN;

  const int rowA0 = (rg * 32 + mn) * 68;
  const int rowA1 = (rg * 32 + 16 + mn) * 68;
  const int rowB0 = (cg * 32 + mn) * 68;
  const int rowB1 = (cg * 32 + 16 + mn) * 68;

  for (int l0 = 0; l0 < L_LEN; l0 += 64) {
    // stage pA: 128x64 floats = 2048 float4; pB: 64x64 = 1024 float4
    for (int rep = 0; rep < 8; ++rep) {
      int idx = rep * 256 + t;        // 0..2047
      int row = idx >> 4;             // 0..127
      int c4  = (idx & 15) * 4;       // 0..60
      *(float4*)&pA[row * 68 + c4] = *(const float4*)&XA[(size_t)row * L_LEN + l0 + c4];
    }
    for (int rep = 0; rep < 4; ++rep) {
      int idx = rep * 256 + t;        // 0..1023
      int row = idx >> 4;             // 0..63
      int c4  = (idx & 15) * 4;
      *(float4*)&pB[row * 68 + c4] = *(const float4*)&XB[(size_t)row * L_LEN + l0 + c4];
    }
    __syncthreads();
    for (int kk = 0; kk < 64; kk += 4) {
      v2f a0 = *(const v2f*)&pA[rowA0 + kk + k0];
      v2f a1 = *(const v2f*)&pA[rowA1 + kk + k0];
      v2f b0 = *(const v2f*)&pB[rowB0 + kk + k0];
      v2f b1 = *(const v2f*)&pB[rowB1 + kk + k0];
      acc00 = wmma4(a0, b0, acc00);
      acc01 = wmma4(a0, b1, acc01);
      acc10 = wmma4(a1, b0, acc10);
      acc11 = wmma4(a1, b1, acc11);
    }
    __syncthreads();
  }

  const int row0 = ti * 128 + rg * 32 + 8 * half;
  const int col0 = tj * 64 + cg * 32 + mn;
  for (int r = 0; r < 8; ++r) {
    unsafeAtomicAdd(&S[(row0 + r) * C_CH + col0],           acc00[r]);
    unsafeAtomicAdd(&S[(row0 + r) * C_CH + col0 + 16],      acc01[r]);
    unsafeAtomicAdd(&S[(row0 + 16 + r) * C_CH + col0],      acc10[r]);
    unsafeAtomicAdd(&S[(row0 + 16 + r) * C_CH + col0 + 16], acc11[r]);
  }
}

// sigma = S/m - mu mu^T + eps*I
__global__ void sigma_kernel(const float* __restrict__ S, const float* __restrict__ SUM,
                             float* __restrict__ SIG) {
  const int i = blockIdx.x, j = threadIdx.x;
  const float invm = 1.0f / (float)M_TOT;
  float mu_i = SUM[i] * invm, mu_j = SUM[j] * invm;
  float v = S[i * C_CH + j] * invm - mu_i * mu_j;
  if (i == j) v += EPSV;
  SIG[i * C_CH + j] = v;
}

// trace + scalars: SCAL[0] = 1/tr, SCAL[1] = rsqrt(tr)
__global__ void trace_kernel(const float* __restrict__ SIG, float* __restrict__ SCAL) {
  const int t = threadIdx.x;
  __shared__ float red[256];
  red[t] = SIG[t * (C_CH + 1)];
  __syncthreads();
  for (int off = 128; off > 0; off >>= 1) {
    if (t < off) red[t] += red[t + off];
    __syncthreads();
  }
  if (t == 0) {
    float s = red[0];
    SCAL[0] = 1.0f / s;
    SCAL[1] = rsqrtf(s);
  }
}

__global__ void initYZ_kernel(const float* __restrict__ SIG, const float* __restrict__ SCAL,
                              float* __restrict__ Y, float* __restrict__ Z) {
  int idx = blockIdx.x * 256 + threadIdx.x;
  int i = idx >> 8, j = idx & 255;
  Y[idx] = SIG[idx] * SCAL[0];
  Z[idx] = (i == j) ? 1.0f : 0.0f;
}

// ------------------------------- generic 256x256x256 WMMA GEMM (NS iters)
// mode 0: D = 1.5*I - 0.5*(A@B)     mode 1: D = A@B
__global__ void gemm256_kernel(const float* __restrict__ A, const float* __restrict__ B,
                               float* __restrict__ D, int mode) {
  const int ti = blockIdx.x >> 2, tj = blockIdx.x & 3;
  const int t = threadIdx.x, lane = t & 31, wv = t >> 5;
  const int half = lane >> 4, mn = lane & 15, k0 = half * 2;
  const int si = wv & 3, sjB = (wv >> 2) * 2;
  const int rowA  = ti * 64 + si * 16 + mn;
  const int colB0 = tj * 64 + sjB * 16 + mn;
  const int colB1 = colB0 + 16;

  v8f acc0 = {}, acc1 = {};
  for (int k = 0; k < 256; k += 4) {
    v2f a = *(const v2f*)&A[rowA * 256 + k + k0];
    v2f b0, b1;
    b0.x = B[(k + k0) * 256 + colB0];
    b0.y = B[(k + k0 + 1) * 256 + colB0];
    b1.x = B[(k + k0) * 256 + colB1];
    b1.y = B[(k + k0 + 1) * 256 + colB1];
    acc0 = wmma4(a, b0, acc0);
    acc1 = wmma4(a, b1, acc1);
  }
  for (int r = 0; r < 8; ++r) {
    int row = ti * 64 + si * 16 + r + 8 * half;
    float v0 = acc0[r], v1 = acc1[r];
    if (mode == 0) {
      v0 = ((row == colB0) ? 1.5f : 0.0f) - 0.5f * v0;
      v1 = ((row == colB1) ? 1.5f : 0.0f) - 0.5f * v1;
    }
    D[row * 256 + colB0] = v0;
    D[row * 256 + colB1] = v1;
  }
}

// b[c] = rsqrt(tr) * dot(Z[c,:], mu)
__global__ void bvec_kernel(const float* __restrict__ Z, const float* __restrict__ SUM,
                            const float* __restrict__ SCAL, float* __restrict__ Bv) {
  const int c = threadIdx.x;
  const float invm = 1.0f / (float)M_TOT;
  float acc = 0.0f;
  for (int j = 0; j < 256; ++j) acc += Z[c * 256 + j] * (SUM[j] * invm);
  Bv[c] = SCAL[1] * acc;
}

// ---------------------------------------- apply: Out = rsqs*(Z @ X[n]) - b
// Grid: x = 128 l-tiles (32 cols), y = 64 batches. LDS panel [256 x 32] pad 36.
__global__ void apply_kernel(const float* __restrict__ X, const float* __restrict__ Z,
                             const float* __restrict__ Bv, const float* __restrict__ SCAL,
                             float* __restrict__ Out) {
  const int l0 = blockIdx.x * 32;
  const int n  = blockIdx.y;
  const int t = threadIdx.x, lane = t & 31, wv = t >> 5;
  const int half = lane >> 4, mn = lane & 15, k0 = half * 2;

  __shared__ float pX[256 * 36];
  const float* Xn = X + ((size_t)n * C_CH) * L_LEN;
  for (int rep = 0; rep < 8; ++rep) {
    int idx = rep * 256 + t;   // 0..2047
    int row = idx >> 3;        // 0..255
    int c4  = (idx & 7) * 4;   // 0..28
    *(float4*)&pX[row * 36 + c4] = *(const float4*)&Xn[(size_t)row * L_LEN + l0 + c4];
  }
  __syncthreads();

  v8f acc00 = {}, acc01 = {}, acc10 = {}, acc11 = {};
  for (int k = 0; k < 256; k += 4) {
    v2f a0 = *(const v2f*)&Z[((wv * 2 + 0) * 16 + mn) * 256 + k + k0];
    v2f a1 = *(const v2f*)&Z[((wv * 2 + 1) * 16 + mn) * 256 + k + k0];
    v2f b0, b1;
    b0.x = pX[(k + k0) * 36 + mn];
    b0.y = pX[(k + k0 + 1) * 36 + mn];
    b1.x = pX[(k + k0) * 36 + 16 + mn];
    b1.y = pX[(k + k0 + 1) * 36 + 16 + mn];
    acc00 = wmma4(a0, b0, acc00);
    acc01 = wmma4(a0, b1, acc01);
    acc10 = wmma4(a1, b0, acc10);
    acc11 = wmma4(a1, b1, acc11);
  }

  const float rsqs = SCAL[1];
  for (int r = 0; r < 8; ++r) {
    int row0 = (wv * 2 + 0) * 16 + r + 8 * half;
    int row1 = (wv * 2 + 1) * 16 + r + 8 * half;
    int col0 = l0 + mn;
    float bb0 = Bv[row0], bb1 = Bv[row1];
    Out[((size_t)(n * C_CH + row0)) * L_LEN + col0]      = rsqs * acc00[r] - bb0;
    Out[((size_t)(n * C_CH + row0)) * L_LEN + col0 + 16] = rsqs * acc01[r] - bb0;
    Out[((size_t)(n * C_CH + row1)) * L_LEN + col0]      = rsqs * acc10[r] - bb1;
    Out[((size_t)(n * C_CH + row1)) * L_LEN + col0 + 16] = rsqs * acc11[r] - bb1;
  }
}

// ---------------------------------------------------------------- launcher
extern "C" void kernel_launch(void* const* d_in, const int* in_sizes, int n_in,
                              void* d_out, int out_size, void* d_ws, size_t ws_size,
                              hipStream_t stream) {
  const float* X = (const float*)d_in[0];
  float* out = (float*)d_out;
  float* W = (float*)d_ws;

  const int MAT = C_CH * C_CH;   // 65536 floats
  float* S    = W;               // raw second moment
  float* SIG  = W + 1 * MAT;     // covariance
  float* Y    = W + 2 * MAT;     // NS Y
  float* Z    = W + 3 * MAT;     // NS Z
  float* T    = W + 4 * MAT;     // NS T
  float* Y2   = W + 5 * MAT;
  float* Z2   = W + 6 * MAT;
  float* SUM  = W + 7 * MAT;     // 256
  float* Bv   = SUM + 256;       // 256
  float* SCAL = Bv + 256;        // 2 scalars

  zero_kernel<<<MAT / 256, 256, 0, stream>>>(S, MAT);
  chan_sum_kernel<<<256, 256, 0, stream>>>(X, SUM);
  cov_kernel<<<dim3(8, 64), 256, 0, stream>>>(X, S);
  sigma_kernel<<<256, 256, 0, stream>>>(S, SUM, SIG);
  trace_kernel<<<1, 256, 0, stream>>>(SIG, SCAL);
  initYZ_kernel<<<MAT / 256, 256, 0, stream>>>(SIG, SCAL, Y, Z);

  float* Yc = Y;  float* Zc = Z;
  float* Yn = Y2; float* Zn = Z2;
  for (int it = 0; it < NS_ITERS; ++it) {
    gemm256_kernel<<<16, 256, 0, stream>>>(Zc, Yc, T, 0);   // T = 1.5I - 0.5*Z@Y
    gemm256_kernel<<<16, 256, 0, stream>>>(Yc, T, Yn, 1);   // Y' = Y@T
    gemm256_kernel<<<16, 256, 0, stream>>>(T, Zc, Zn, 1);   // Z' = T@Z
    float* tmp;
    tmp = Yc; Yc = Yn; Yn = tmp;
    tmp = Zc; Zc = Zn; Zn = tmp;
  }

  bvec_kernel<<<1, 256, 0, stream>>>(Zc, SUM, SCAL, Bv);
  apply_kernel<<<dim3(L_LEN / 32, N_B), 256, 0, stream>>>(X, Zc, Bv, SCAL, out);
}